// TransformerBlock_7722351198653
// MI455X (gfx1250) — compile-verified
//
#include <hip/hip_runtime.h>
#include <hip/hip_bf16.h>
#include <math.h>

// ---------------- model constants ----------------
#define S    16
#define D    1024
#define Hf   1024
#define NE   32
#define KTOP 4
#define NH   16
#define NKV  4
#define HD   64
#define LIM  7.0f
#define EPSV 1e-5f

// ---------------- workspace layout (floats) ----------------
#define OFF_HX   0                       // 16*1024
#define OFF_Q    (OFF_HX  + S*D)         // 16*1024
#define OFF_K    (OFF_Q   + S*D)         // 16*256
#define OFF_V    (OFF_K   + S*NKV*HD)    // 16*256
#define OFF_ATT  (OFF_V   + S*NKV*HD)    // 16*1024
#define OFF_H    (OFF_ATT + S*D)         // 16*1024
#define OFF_XF   (OFF_H   + S*D)         // 16*1024
#define OFF_EW   (OFF_XF  + S*D)         // 64
#define OFF_IDX  (OFF_EW  + S*KTOP)      // 64 ints
#define OFF_H1   (OFF_IDX + S*KTOP)      // 64*2048
#define OFF_ACT  (OFF_H1  + S*KTOP*2*Hf) // 64*1024
// total ~ 286848 floats ~= 1.15 MB

typedef float v2f __attribute__((ext_vector_type(2)));
typedef float v8f __attribute__((ext_vector_type(8)));

// ============================================================
// fp32 WMMA 16x16 tile partial GEMM over K range [k0, k0+KCHUNK):
//   C += A[16, k-range] * W[n0:n0+16, k-range]^T
// A row-major (lda), W row-major (ldw). Fragment layouts per
// CDNA5 ISA 7.12.2 (wave32, EXEC all ones).
// k0 is wave-uniform (readfirstlane'd by callers) -> scalar base;
// KCHUNK is a compile-time constant -> unrolled, clause-pipelined loop.
// ============================================================
template <int KCHUNK>
__device__ inline v8f wmma_ktile_f32(const float* __restrict__ A, int lda,
                                     const float* __restrict__ W, int ldw,
                                     int n0, int k0, int lane) {
  v8f c = {};
  const int ln = lane & 15;          // M for A-frag, N for B-frag
  const int ko = (lane >> 4) << 1;   // lanes 0-15 -> K{0,1}, lanes 16-31 -> K{2,3}
  const float* arow = A + (size_t)ln * lda + k0 + ko;
  const float* wrow = W + (size_t)(n0 + ln) * ldw + k0 + ko;
#pragma unroll 16
  for (int kk = 0; kk < KCHUNK; kk += 4) {
    float2 af = *(const float2*)(arow + kk);
    float2 bf = *(const float2*)(wrow + kk);
    v2f a; a.x = af.x; a.y = af.y;
    v2f b; b.x = bf.x; b.y = bf.y;
    c = __builtin_amdgcn_wmma_f32_16x16x4_f32(false, a, false, b, (short)0, c,
                                              false, false);
  }
  return c;
}

// Cross-wave reduce of 4 partial v8f accumulators through LDS.
// Returns summed tile in threads 0..31 (others hold garbage).
__device__ inline v8f ksplit_reduce(v8f c, float* red /* [4][8][32] */, int wave) {
  const int lane = threadIdx.x & 31;
#pragma unroll
  for (int r = 0; r < 8; ++r) red[(wave * 8 + r) * 32 + lane] = c[r];
  __syncthreads();
  v8f s = {};
  if (threadIdx.x < 32) {
#pragma unroll
    for (int r = 0; r < 8; ++r)
      s[r] = red[(0 * 8 + r) * 32 + lane] + red[(1 * 8 + r) * 32 + lane] +
             red[(2 * 8 + r) * 32 + lane] + red[(3 * 8 + r) * 32 + lane];
  }
  return s;
}

// ============================================================
// RMSNorm: out[t][:] = x[t][:] * rsqrt(mean(x^2)+eps) * w
// grid = 16 blocks, block = 256
// ============================================================
__global__ void rmsnorm_kernel(const float* __restrict__ x,
                               const float* __restrict__ w,
                               float* __restrict__ out) {
  __shared__ float red[256];
  const int t = blockIdx.x, tid = threadIdx.x;
  float s = 0.f;
  for (int i = tid; i < D; i += 256) { float v = x[t * D + i]; s += v * v; }
  red[tid] = s; __syncthreads();
  for (int o = 128; o > 0; o >>= 1) {
    if (tid < o) red[tid] += red[tid + o];
    __syncthreads();
  }
  const float r = rsqrtf(red[0] / (float)D + EPSV);
  for (int i = tid; i < D; i += 256) out[t * D + i] = x[t * D + i] * r * w[i];
}

// ============================================================
// QKV projection via K-split WMMA. grid.x = 96 tiles (64 Q, 16 K, 16 V),
// block = 128 (4 waves; wave w reduces K chunk [w*256,(w+1)*256)).
// ============================================================
__global__ void qkv_kernel(const float* __restrict__ hx,
                           const float* __restrict__ wq, const float* __restrict__ bq,
                           const float* __restrict__ wk, const float* __restrict__ bk,
                           const float* __restrict__ wv, const float* __restrict__ bv,
                           float* __restrict__ q, float* __restrict__ k,
                           float* __restrict__ v) {
  __shared__ float red[4 * 8 * 32];
  const int lane = threadIdx.x & 31;
  // wave index as an SGPR: scalar K base, full-EXEC WMMA loop
  const int wave = __builtin_amdgcn_readfirstlane(threadIdx.x >> 5);
  const int tt = blockIdx.x;
  const float* W; const float* bias; float* out; int n0, ldo;
  if (tt < 64)      { W = wq; bias = bq; out = q; n0 = tt * 16;        ldo = NH * HD; }
  else if (tt < 80) { W = wk; bias = bk; out = k; n0 = (tt - 64) * 16; ldo = NKV * HD; }
  else              { W = wv; bias = bv; out = v; n0 = (tt - 80) * 16; ldo = NKV * HD; }
  v8f c = wmma_ktile_f32<D / 4>(hx, D, W, D, n0, wave * (D / 4), lane);
  c = ksplit_reduce(c, red, wave);
  if (threadIdx.x < 32) {
    const int ln = lane & 15, mh = (lane >> 4) << 3;
#pragma unroll
    for (int r = 0; r < 8; ++r)
      out[(size_t)(mh + r) * ldo + n0 + ln] = c[r] + bias[n0 + ln];
  }
}

// ============================================================
// RoPE in place on q (16x16x64) and k (16x4x64). 10240 work items.
// ============================================================
__global__ void rope_kernel(float* __restrict__ q, float* __restrict__ k,
                            const float* __restrict__ fc,
                            const float* __restrict__ fs) {
  const int i = blockIdx.x * blockDim.x + threadIdx.x;
  if (i < S * NH * 32) {
    const int p = i & 31, hh = (i >> 5) & 15, t = i >> 9;
    const float c = fc[t * 32 + p], s = fs[t * 32 + p];
    float* base = q + t * (NH * HD) + hh * HD + 2 * p;
    const float t1 = base[0], t2 = base[1];
    base[0] = t1 * c - t2 * s; base[1] = t1 * s + t2 * c;
  } else if (i < S * NH * 32 + S * NKV * 32) {
    const int j = i - S * NH * 32;
    const int p = j & 31, hh = (j >> 5) & 3, t = j >> 7;
    const float c = fc[t * 32 + p], s = fs[t * 32 + p];
    float* base = k + t * (NKV * HD) + hh * HD + 2 * p;
    const float t1 = base[0], t2 = base[1];
    base[0] = t1 * c - t2 * s; base[1] = t1 * s + t2 * c;
  }
}

// ============================================================
// Attention per head: WMMA scores (16x16,K=64), softmax+sink in LDS,
// WMMA probs@V (16x64). grid = 16 heads, block = 32 (one wave).
// ============================================================
__global__ void attn_kernel(const float* __restrict__ q,
                            const float* __restrict__ kws,
                            const float* __restrict__ vws,
                            const float* __restrict__ sinks,
                            float* __restrict__ attn) {
  __shared__ float sc[S * S];
  __shared__ float pr[S * S];
  const int h = blockIdx.x, lane = threadIdx.x;
  const int kv = h >> 2;   // G = NH/NKV = 4
  const int ln = lane & 15, ko = (lane >> 4) << 1, mh = (lane >> 4) << 3;

  // scores = Q_h (16x64) x K_h^T
  v8f c = {};
  const float* qrow = q + (size_t)ln * (NH * HD) + h * HD + ko;
  const float* krow = kws + (size_t)ln * (NKV * HD) + kv * HD + ko;
#pragma unroll
  for (int kk = 0; kk < HD; kk += 4) {
    float2 af = *(const float2*)(qrow + kk);
    float2 bf = *(const float2*)(krow + kk);
    v2f a; a.x = af.x; a.y = af.y;
    v2f b; b.x = bf.x; b.y = bf.y;
    c = __builtin_amdgcn_wmma_f32_16x16x4_f32(false, a, false, b, (short)0, c,
                                              false, false);
  }
#pragma unroll
  for (int r = 0; r < 8; ++r) sc[(mh + r) * S + ln] = c[r] * 0.125f;
  __syncthreads();

  // causal softmax with sink logit (WIN=128 > S, never binds)
  if (lane < S) {
    const int i = lane;
    const float snk = sinks[h];
    float m = snk;
    for (int j = 0; j <= i; ++j) m = fmaxf(m, sc[i * S + j]);
    float sum = __expf(snk - m);
    for (int j = 0; j < S; ++j) {
      const float e = (j <= i) ? __expf(sc[i * S + j] - m) : 0.f;
      pr[i * S + j] = e; sum += e;
    }
    const float inv = 1.f / sum;
    for (int j = 0; j < S; ++j) pr[i * S + j] *= inv;
  }
  __syncthreads();

  // attn = probs (16x16) @ V_h (16x64) -> 4 N-tiles
  for (int nt = 0; nt < 4; ++nt) {
    v8f d = {};
#pragma unroll
    for (int ks = 0; ks < S; ks += 4) {
      float2 af = *(const float2*)(&pr[ln * S + ks + ko]);
      v2f a; a.x = af.x; a.y = af.y;
      v2f b;
      b.x = vws[(size_t)(ks + ko)     * (NKV * HD) + kv * HD + nt * 16 + ln];
      b.y = vws[(size_t)(ks + ko + 1) * (NKV * HD) + kv * HD + nt * 16 + ln];
      d = __builtin_amdgcn_wmma_f32_16x16x4_f32(false, a, false, b, (short)0, d,
                                                false, false);
    }
#pragma unroll
    for (int r = 0; r < 8; ++r)
      attn[(size_t)(mh + r) * (NH * HD) + h * HD + nt * 16 + ln] = d[r];
  }
}

// ============================================================
// O projection + residual: h = x + attn @ wo^T + bo  (writes ws h and d_out)
// grid = 64 tiles, block = 128 (4-way K split)
// ============================================================
__global__ void oproj_kernel(const float* __restrict__ attn,
                             const float* __restrict__ wo,
                             const float* __restrict__ bo,
                             const float* __restrict__ x,
                             float* __restrict__ h, float* __restrict__ outf) {
  __shared__ float red[4 * 8 * 32];
  const int lane = threadIdx.x & 31;
  const int wave = __builtin_amdgcn_readfirstlane(threadIdx.x >> 5);
  const int n0 = blockIdx.x * 16;
  const int KD = NH * HD;
  v8f c = wmma_ktile_f32<(NH * HD) / 4>(attn, KD, wo, KD, n0, wave * (KD / 4), lane);
  c = ksplit_reduce(c, red, wave);
  if (threadIdx.x < 32) {
    const int ln = lane & 15, mh = (lane >> 4) << 3;
#pragma unroll
    for (int r = 0; r < 8; ++r) {
      const int m = mh + r;
      const float v = x[(size_t)m * D + n0 + ln] + c[r] + bo[n0 + ln];
      h[(size_t)m * D + n0 + ln] = v;
      outf[(size_t)m * D + n0 + ln] = v;
    }
  }
}

// ============================================================
// Router: logits (32 per token), top-4, softmax weights.
// grid = 16 tokens, block = 32 (lane e computes logit e)
// ============================================================
__global__ void gate_topk_kernel(const float* __restrict__ xf,
                                 const float* __restrict__ gw,
                                 const float* __restrict__ gb,
                                 float* __restrict__ ew, int* __restrict__ idx) {
  __shared__ float gs[NE];
  const int t = blockIdx.x, lane = threadIdx.x;
  const float* xr = xf + (size_t)t * D;
  const float* wr = gw + (size_t)lane * D;
  float g = gb[lane];
  for (int k0 = 0; k0 < D; k0 += 4) {
    float4 a = *(const float4*)(wr + k0);
    float4 b = *(const float4*)(xr + k0);
    g = fmaf(a.x, b.x, g); g = fmaf(a.y, b.y, g);
    g = fmaf(a.z, b.z, g); g = fmaf(a.w, b.w, g);
  }
  gs[lane] = g;
  __syncthreads();
  if (lane == 0) {
    int id[KTOP]; float vv[KTOP];
    for (int s = 0; s < KTOP; ++s) {
      int am = 0; float mv = gs[0];
      for (int e = 1; e < NE; ++e)
        if (gs[e] > mv) { mv = gs[e]; am = e; }
      id[s] = am; vv[s] = mv; gs[am] = -1e30f;
    }
    float sum = 0.f, ee[KTOP];
    for (int s = 0; s < KTOP; ++s) { ee[s] = __expf(vv[s] - vv[0]); sum += ee[s]; }
    for (int s = 0; s < KTOP; ++s) {
      ew[t * KTOP + s] = ee[s] / sum;
      idx[t * KTOP + s] = id[s];
    }
  }
}

// ============================================================
// Wave GEMV dot of length 1024, float4 streaming + butterfly reduce.
// ============================================================
__device__ inline float wave_dot1024(const float* __restrict__ w,
                                     const float* __restrict__ x, int lane) {
  float acc = 0.f;
  __builtin_prefetch(w + D, 0, 0);   // stream-ahead hint -> global_prefetch_b8
#pragma unroll
  for (int it = 0; it < 8; ++it) {
    const int k0 = it * 128 + lane * 4;
    float4 a = *(const float4*)(w + k0);
    float4 b = *(const float4*)(x + k0);
    acc = fmaf(a.x, b.x, acc); acc = fmaf(a.y, b.y, acc);
    acc = fmaf(a.z, b.z, acc); acc = fmaf(a.w, b.w, acc);
  }
  for (int o = 16; o > 0; o >>= 1) acc += __shfl_down(acc, o, 32);
  return acc;
}

// ============================================================
// MoE stage 1: h1[pair][row] = w1[e][row] . xf[t] + b1[e][row]
// grid = (2048/8, 64 pairs), block = 256 (8 waves, one row each)
// ============================================================
__global__ void moe_h1_kernel(const float* __restrict__ xf,
                              const float* __restrict__ w1,
                              const float* __restrict__ b1,
                              const int* __restrict__ idx,
                              float* __restrict__ h1) {
  const int p = blockIdx.y, t = p >> 2, s = p & 3;
  const int e = idx[t * KTOP + s];
  const int wave = threadIdx.x >> 5, lane = threadIdx.x & 31;
  const int row = blockIdx.x * 8 + wave;
  const float* wr = w1 + ((size_t)e * (2 * Hf) + row) * D;
  const float acc = wave_dot1024(wr, xf + (size_t)t * D, lane);
  if (lane == 0) h1[(size_t)p * (2 * Hf) + row] = acc + b1[(size_t)e * (2 * Hf) + row];
}

// ============================================================
// Clamped SwiGLU-ish activation (per reference).
// ============================================================
__global__ void moe_act_kernel(const float* __restrict__ h1,
                               float* __restrict__ act) {
  const int i = blockIdx.x * blockDim.x + threadIdx.x;
  if (i < S * KTOP * Hf) {
    const int p = i >> 10, c = i & 1023;
    float g = h1[(size_t)p * (2 * Hf) + 2 * c];
    float l = h1[(size_t)p * (2 * Hf) + 2 * c + 1];
    g = fminf(g, LIM);
    l = fminf(fmaxf(l, -LIM), LIM);
    const float sg = 1.f / (1.f + __expf(-1.702f * g));
    act[(size_t)p * Hf + c] = g * sg * (l + 1.f);
  }
}

// ============================================================
// MoE stage 2: out[t][row] += ew * (w2[e][row] . act[pair] + b2[e][row])
// grid = (1024/8, 64 pairs), block = 256
// ============================================================
__global__ void moe_h2_kernel(const float* __restrict__ act,
                              const float* __restrict__ w2,
                              const float* __restrict__ b2,
                              const int* __restrict__ idx,
                              const float* __restrict__ ew,
                              float* __restrict__ outf) {
  const int p = blockIdx.y, t = p >> 2, s = p & 3;
  const int e = idx[t * KTOP + s];
  const int wave = threadIdx.x >> 5, lane = threadIdx.x & 31;
  const int row = blockIdx.x * 8 + wave;
  const float* wr = w2 + ((size_t)e * D + row) * Hf;
  const float acc = wave_dot1024(wr, act + (size_t)p * Hf, lane);
  if (lane == 0) {
    const float v = (acc + b2[(size_t)e * D + row]) * ew[t * KTOP + s];
    atomicAdd(&outf[(size_t)t * D + row], v);
  }
}

// ============================================================
extern "C" void kernel_launch(void* const* d_in, const int* in_sizes, int n_in,
                              void* d_out, int out_size, void* d_ws, size_t ws_size,
                              hipStream_t stream) {
  const float* x   = (const float*)d_in[0];
  const float* fc  = (const float*)d_in[1];
  const float* fs  = (const float*)d_in[2];
  const float* anw = (const float*)d_in[3];
  const float* wq  = (const float*)d_in[4];
  const float* bq  = (const float*)d_in[5];
  const float* wk  = (const float*)d_in[6];
  const float* bk  = (const float*)d_in[7];
  const float* wv  = (const float*)d_in[8];
  const float* bv  = (const float*)d_in[9];
  const float* wo  = (const float*)d_in[10];
  const float* bo  = (const float*)d_in[11];
  const float* snk = (const float*)d_in[12];
  const float* fnw = (const float*)d_in[13];
  const float* gw  = (const float*)d_in[14];
  const float* gb  = (const float*)d_in[15];
  const float* w1  = (const float*)d_in[16];
  const float* b1  = (const float*)d_in[17];
  const float* w2  = (const float*)d_in[18];
  const float* b2  = (const float*)d_in[19];

  float* ws   = (float*)d_ws;
  float* hx   = ws + OFF_HX;
  float* q    = ws + OFF_Q;
  float* kx   = ws + OFF_K;
  float* vx   = ws + OFF_V;
  float* attn = ws + OFF_ATT;
  float* h    = ws + OFF_H;
  float* xf   = ws + OFF_XF;
  float* ew   = ws + OFF_EW;
  int*   idx  = (int*)(ws + OFF_IDX);
  float* h1   = ws + OFF_H1;
  float* act  = ws + OFF_ACT;
  float* outf = (float*)d_out;

  rmsnorm_kernel<<<S, 256, 0, stream>>>(x, anw, hx);
  qkv_kernel<<<96, 128, 0, stream>>>(hx, wq, bq, wk, bk, wv, bv, q, kx, vx);
  rope_kernel<<<40, 256, 0, stream>>>(q, kx, fc, fs);
  attn_kernel<<<NH, 32, 0, stream>>>(q, kx, vx, snk, attn);
  oproj_kernel<<<64, 128, 0, stream>>>(attn, wo, bo, x, h, outf);
  rmsnorm_kernel<<<S, 256, 0, stream>>>(h, fnw, xf);
  gate_topk_kernel<<<S, 32, 0, stream>>>(xf, gw, gb, ew, idx);
  moe_h1_kernel<<<dim3(2 * Hf / 8, S * KTOP), 256, 0, stream>>>(xf, w1, b1, idx, h1);
  moe_act_kernel<<<(S * KTOP * Hf + 255) / 256, 256, 0, stream>>>(h1, act);
  moe_h2_kernel<<<dim3(D / 8, S * KTOP), 256, 0, stream>>>(act, w2, b2, idx, ew, outf);
}